// PowerSpectrum_2843268350373
// MI455X (gfx1250) — compile-verified
//
#include <hip/hip_runtime.h>

// PowerSpectrum densified: out[s,n,l,q,p] = (1/sqrt(2l+1)) * sum_m v[s,n,l,m,q]*v[s,n,l,m,p]
// v: [S, N, L, M, Q] fp32, m masked to first 2l+1 channels.  S=4 N=2000 L=4 M=7 Q=64.
//
// Memory-bound (0.58 GB traffic, 1.8 GFLOP). Strategy: one wave per (s,n,l),
// fp32 WMMA 16x16x4 for the Gram tiles, non-temporal stores for the 524 MB output.

typedef float v2f __attribute__((ext_vector_type(2)));
typedef float v8f __attribute__((ext_vector_type(8)));

#define L_DIM 4
#define M_DIM 7
#define Q_DIM 64

__global__ __launch_bounds__(128) void powerspectrum_wmma_kernel(
    const float* __restrict__ vals, float* __restrict__ out) {
  const int b    = blockIdx.x;          // flattened (s, n)
  const int l    = threadIdx.x >> 5;    // wave id == l  (4 waves per block)
  const int lane = threadIdx.x & 31;
  const int r    = lane & 15;           // row/col within 16-wide tile
  const int h    = lane >> 4;           // lane-half: selects K pair {2h, 2h+1}

  const float* __restrict__ src = vals + ((size_t)b * L_DIM + l) * (M_DIM * Q_DIM);
  float* __restrict__       dst = out  + ((size_t)b * L_DIM + l) * (Q_DIM * Q_DIM);

  const int   mvalid = 2 * l + 1;                 // valid m channels (<= 7)
  const float cg     = rsqrtf((float)(2 * l + 1)); // 1/sqrt(2l+1)

  // Shared A/B fragments. ISA layout (32-bit, 16x4 A / 4x16 B): lane (r,h),
  // VGPR j  <->  k = 2h + j.  frag[base][ks][j] = v[4*ks + 2h + j][16*base + r],
  // zero for masked/padded m (m >= 2l+1, incl. the m==7 pad).
  v2f frag[4][2];
#pragma unroll
  for (int base = 0; base < 4; ++base) {
#pragma unroll
    for (int ks = 0; ks < 2; ++ks) {
#pragma unroll
      for (int j = 0; j < 2; ++j) {
        const int m = 4 * ks + 2 * h + j;
        float x = 0.0f;
        if (m < mvalid)                     // predicated load; EXEC restored
          x = src[m * Q_DIM + base * 16 + r];
        frag[base][ks][j] = x;
      }
    }
  }

  // 16 output tiles of 16x16; two K=4 WMMA steps each (m 0..3, m 4..7).
  // Fold cg into the A-side operand only -> D = cg * A^T A.
#pragma unroll
  for (int qb = 0; qb < 4; ++qb) {
    const v2f a0 = frag[qb][0] * cg;
    const v2f a1 = frag[qb][1] * cg;
#pragma unroll
    for (int pb = 0; pb < 4; ++pb) {
      v8f acc = {};
      acc = __builtin_amdgcn_wmma_f32_16x16x4_f32(
          /*neg_a=*/false, a0, /*neg_b=*/false, frag[pb][0],
          /*c_mod=*/(short)0, acc, /*reuse_a=*/false, /*reuse_b=*/false);
      acc = __builtin_amdgcn_wmma_f32_16x16x4_f32(
          /*neg_a=*/false, a1, /*neg_b=*/false, frag[pb][1],
          /*c_mod=*/(short)0, acc, /*reuse_a=*/false, /*reuse_b=*/false);

      // D layout: VGPR i, lane (r,h) -> element (q = 16*qb + i + 8h, p = 16*pb + r).
      // Output written once, never re-read -> non-temporal stores.
#pragma unroll
      for (int i = 0; i < 8; ++i) {
        const size_t q = (size_t)(qb * 16 + i + 8 * h);
        __builtin_nontemporal_store(acc[i], dst + q * Q_DIM + pb * 16 + r);
      }
    }
  }
}

extern "C" void kernel_launch(void* const* d_in, const int* in_sizes, int n_in,
                              void* d_out, int out_size, void* d_ws, size_t ws_size,
                              hipStream_t stream) {
  const float* vals = (const float*)d_in[0];
  float* out = (float*)d_out;
  // in_sizes[0] = S*N*L*M*Q ; one block per (s,n), 4 waves (one per l) per block.
  const int num_sn = in_sizes[0] / (L_DIM * M_DIM * Q_DIM);
  powerspectrum_wmma_kernel<<<num_sn, 128, 0, stream>>>(vals, out);
}